// CentreDependentPooling2D_58961311039819
// MI455X (gfx1250) — compile-verified
//
#include <hip/hip_runtime.h>
#include <stdint.h>

// CentreDependentPooling2D for MI455X (gfx1250, wave32).
// Band-dependent max-pool, stride 2, SAME (-inf) padding.
//   k(band) = 2 + 6*band, pad_before(band) = 3*band, bands by radius from (128,128).
// Strategy: 16x16 output tile per workgroup, 4 channels (one float4) per workgroup.
//   Stage 0: async-load input region (<=62x62 px, 16B/px) into LDS (halo = -inf).
//   Stage 1a: pairwise row maxes at both parities (any even-k vertical window
//             = k/2 pair maxes).
//   Stage 1b: vertical maxes for each band present in the tile (<=3 bands).
//   Stage 2: per-pixel exact band -> horizontal max -> b128 store.

#define Hh   512
#define Ww   512
#define Cc   64
#define OUT  256
#define CTR  128
#define TO   16     // output tile edge
#define SX   63     // padded x-stride inside LDS tiles (in float4 units)
#define HALF4 3906  // 62 * 63 float4 slots per LDS plane

#if __has_builtin(__builtin_amdgcn_global_load_async_to_lds_b128)
#define HAVE_ASYNC_LDS 1
#else
#define HAVE_ASYNC_LDS 0
#endif

typedef int v4i __attribute__((vector_size(4 * sizeof(int))));
typedef __attribute__((address_space(1))) v4i* as1_v4i_ptr;
typedef __attribute__((address_space(3))) v4i* as3_v4i_ptr;

typedef float f4 __attribute__((ext_vector_type(4)));

__device__ __forceinline__ int band_of(int d2) {
  return (d2 >= 3600) + (d2 >= 5625) + (d2 >= 8100) + (d2 >= 11025) + (d2 >= 14400);
}

__device__ __forceinline__ f4 vmax4(f4 a, f4 b) {
  f4 r;
  r.x = fmaxf(a.x, b.x);
  r.y = fmaxf(a.y, b.y);
  r.z = fmaxf(a.z, b.z);
  r.w = fmaxf(a.w, b.w);
  return r;
}

__device__ __forceinline__ void async_ld16(const float* g, f4* l) {
#if HAVE_ASYNC_LDS
  // global (AS1) -> LDS (AS3), 128-bit chunk, tracked by ASYNCcnt.
  __builtin_amdgcn_global_load_async_to_lds_b128(
      (as1_v4i_ptr)(uintptr_t)g,
      (as3_v4i_ptr)(uint32_t)(uintptr_t)l,
      0, 0);
#else
  const f4* gs = (const f4*)g;
  *l = *gs;
#endif
}

__global__ __launch_bounds__(256, 1)
void cdp_pool_kernel(const float* __restrict__ x, float* __restrict__ out) {
  __shared__ __align__(16) f4 smem[2 * HALF4];
  f4* sIn = smem;          // input tile  [row][x]  (row-stride SX float4s)
  f4* sP  = smem + HALF4;  // pair maxes: parity0 rows 0..30, parity1 rows 31..61
  f4* sV  = smem;          // vertical maxes [lvl][oy][x] — aliases sIn (dead)

  const int t    = (int)threadIdx.x;
  const int tx   = (int)blockIdx.x >> 4;
  const int cblk = (int)blockIdx.x & 15;
  const int ty   = (int)blockIdx.y;
  const int b    = (int)blockIdx.z;

  const int oy0 = ty * TO, ox0 = tx * TO;

  // Band range over the tile (exact: clamp center into tile / farthest corner).
  int ny = CTR; if (ny < oy0) ny = oy0; if (ny > oy0 + TO - 1) ny = oy0 + TO - 1;
  int nx = CTR; if (nx < ox0) nx = ox0; if (nx > ox0 + TO - 1) nx = ox0 + TO - 1;
  int dyn = ny - CTR, dxn = nx - CTR;
  const int d2min = dyn * dyn + dxn * dxn;
  int a1 = oy0 - CTR; if (a1 < 0) a1 = -a1;
  int a2 = oy0 + TO - 1 - CTR; if (a2 < 0) a2 = -a2;
  int fy = a1 > a2 ? a1 : a2;
  a1 = ox0 - CTR; if (a1 < 0) a1 = -a1;
  a2 = ox0 + TO - 1 - CTR; if (a2 < 0) a2 = -a2;
  int fx = a1 > a2 ? a1 : a2;
  const int d2max = fy * fy + fx * fx;

  const int blo  = band_of(d2min);
  const int bhi  = band_of(d2max);      // bhi - blo <= 2 (tile diameter < 2 band widths)
  const int khi  = 2 + 6 * bhi;
  const int pbhi = 3 * bhi;
  const int RH   = 30 + khi;            // input region edge (rows == cols), <= 62
  const int ry0  = 2 * oy0 - pbhi;
  const int cx0  = 2 * ox0 - pbhi;

  const float NEG = -__builtin_inff();
  const f4 NEG4 = {NEG, NEG, NEG, NEG};

  // Stage 0a: fill tile (incl. halo) with -inf => SAME padding semantics.
  for (int i = t; i < RH * SX; i += 256) sIn[i] = NEG4;
  __syncthreads();

  // Stage 0b: async gather of in-bounds 16B chunks (4 channels) into LDS.
  const float* gbase = x + ((size_t)b * Hh * Ww) * Cc + (size_t)cblk * 4;
  for (int i = t; i < RH * RH; i += 256) {
    int r  = i / RH;
    int xx = i - r * RH;
    int gy = ry0 + r, gx = cx0 + xx;
    if ((unsigned)gy < (unsigned)Hh && (unsigned)gx < (unsigned)Ww) {
      const float* g = gbase + ((size_t)gy * Ww + gx) * Cc;
      async_ld16(g, sIn + r * SX + xx);
    }
  }
  asm volatile("s_wait_asynccnt 0" ::: "memory");
  __syncthreads();

  // Stage 1a: pairwise vertical maxes at both parities.
  //   parity0: P[j]    = max(row 2j,   row 2j+1),  j < RH/2
  //   parity1: P[31+j] = max(row 2j+1, row 2j+2),  j < RH/2 - 1
  const int NP0 = RH >> 1;
  const int NP1 = NP0 - 1;
  for (int i = t; i < (NP0 + NP1) * RH; i += 256) {
    int xx = i % RH;
    int pj = i / RH;
    int r0, dst;
    if (pj < NP0) { r0 = 2 * pj;                      dst = pj;     }
    else          { int j = pj - NP0; r0 = 2 * j + 1;  dst = 31 + j; }
    sP[dst * SX + xx] = vmax4(sIn[r0 * SX + xx], sIn[(r0 + 1) * SX + xx]);
  }
  __syncthreads();

  // Stage 1b: vertical window maxes per band level present in this tile.
  //   window rows [s, s+k), s = 2*oy_local + 3*(bhi - lvl), k = 2+6*lvl (even)
  //   => k/2 consecutive pair-maxes of parity (s & 1).
  const int nlev = bhi - blo + 1;
  for (int i = t; i < nlev * TO * RH; i += 256) {
    int xx  = i % RH;
    int t3  = i / RH;
    int oyl = t3 & (TO - 1);
    int l   = t3 >> 4;
    int lvl = blo + l;
    int k   = 2 + 6 * lvl;
    int s   = 2 * oyl + 3 * (bhi - lvl);
    int base = (s & 1) ? 31 : 0;
    int j0   = s >> 1;
    f4 m = NEG4;
    int kh = k >> 1;
    for (int q = 0; q < kh; ++q)
      m = vmax4(m, sP[(base + j0 + q) * SX + xx]);
    sV[(l * TO + oyl) * SX + xx] = m;
  }
  __syncthreads();

  // Stage 2: per-pixel exact band, horizontal max, one float4 store per thread.
  {
    int oxl = t & (TO - 1);
    int oyl = t >> 4;
    int oy = oy0 + oyl, ox = ox0 + oxl;
    int dy = oy - CTR, dx = ox - CTR;
    int lvl = band_of(dy * dy + dx * dx);     // in [blo, bhi]
    int l   = lvl - blo;
    int k   = 2 + 6 * lvl;
    int xs  = 2 * oxl + 3 * (bhi - lvl);
    f4 m = NEG4;
    const f4* row = sV + (l * TO + oyl) * SX + xs;
    for (int q = 0; q < k; ++q)
      m = vmax4(m, row[q]);
    f4* dst = (f4*)(out + (((size_t)b * OUT + oy) * OUT + ox) * Cc + (size_t)cblk * 4);
    *dst = m;
  }
}

extern "C" void kernel_launch(void* const* d_in, const int* in_sizes, int n_in,
                              void* d_out, int out_size, void* d_ws, size_t ws_size,
                              hipStream_t stream) {
  (void)in_sizes; (void)n_in; (void)d_ws; (void)ws_size; (void)out_size;
  const float* x = (const float*)d_in[0];
  float* out = (float*)d_out;
  // grid.x: cblk (fast, for L2 line reuse across channel blocks) x tile_x
  dim3 grid(16 * 16, 16, 4);
  dim3 block(256);
  cdp_pool_kernel<<<grid, block, 0, stream>>>(x, out);
}